// W_Transform_45938970198411
// MI455X (gfx1250) — compile-verified
//
#include <hip/hip_runtime.h>

typedef __attribute__((ext_vector_type(2))) float v2f;
typedef __attribute__((ext_vector_type(8))) float v8f;

// Haar level-1 along S:  x(64,4096,256)f32 -> out(64,2048,512)f32
//   cA=(x0+x1)*c, cD=(x0-x1)*c, interleaved per feature.  c = 1/sqrt(2).
// Bandwidth-bound (512MB @ 23.3TB/s ~ 22us). Butterfly is folded into a
// constant A operand of V_WMMA_F32_16X16X4_F32 so the matrix pipe does the
// math while the wave streams with fully coalesced NT loads/stores.

#ifndef INV_SQRT2F
#define INV_SQRT2F 0.70710678118654752440f
#endif

constexpr int BDIM  = 64;
constexpr int S     = 4096;
constexpr int F     = 256;
constexpr int PAIRS = S / 2;   // 2048
constexpr int ROWS  = 8;       // pair-rows per block

__global__ __launch_bounds__(256)
void haar_wmma_kernel(const float* __restrict__ x, float* __restrict__ out) {
  const int tid  = threadIdx.x;
  const int lane = tid & 31;
  const int wv   = tid >> 5;      // wave id 0..7 -> feature chunk
  const int f0   = wv * 32;       // 32 features per wave per row

  // Constant Haar operator in A (16x4, f32):
  //   row0=(c, c,0,0)  row1=(c,-c,0,0)  row2=(0,0,c, c)  row3=(0,0,c,-c)
  // Per the ISA 16x4 f32 A layout (VGPR0=K0|K2, VGPR1=K1|K3 split at lane 16):
  //   lane 0 & 18 -> (c, c);  lane 1 & 19 -> (c,-c);  all other lanes 0.
  const float c   = INV_SQRT2F;
  const bool  sA  = (lane == 0) | (lane == 18);
  const bool  sB  = (lane == 1) | (lane == 19);
  v2f A;
  A.x = (sA | sB) ? c : 0.0f;
  A.y = sA ? c : (sB ? -c : 0.0f);

  const int block = blockIdx.x;              // 0 .. BDIM*PAIRS/ROWS - 1
  const int b     = block / (PAIRS / ROWS);
  const int i0    = (block % (PAIRS / ROWS)) * ROWS;

  const float* xb = x   + (size_t)b * S * F;
  float*       ob = out + (size_t)b * PAIRS * (2 * F);

#pragma unroll
  for (int r = 0; r < ROWS; ++r) {
    const int i = i0 + r;

    // B matrix (4x16) arrives layout-correct from two plain coalesced loads:
    //   VGPR0: lane l = x0[f0+l]  (K0 = chunk p on lanes 0-15, K2 = chunk q on 16-31)
    //   VGPR1: lane l = x1[f0+l]  (K1 / K3)
    const float* row0 = xb + (size_t)(2 * i) * F + f0 + lane;
    const float* row1 = row0 + F;
    v2f Bm;
    Bm.x = __builtin_nontemporal_load(row0);   // streaming: no L2 reuse
    Bm.y = __builtin_nontemporal_load(row1);

    v8f Cz = {};
    // D rows: 0=cA_p 1=cD_p 2=cA_q 3=cD_q (on lanes 0-15 of D[0..3])
    v8f D = __builtin_amdgcn_wmma_f32_16x16x4_f32(
        /*neg_a=*/false, A, /*neg_b=*/false, Bm,
        /*c_mod=*/(short)0, Cz, /*reuse_a=*/false, /*reuse_b=*/false);

    // Move chunk-q results (lanes 0-15 of D2/D3) up to lanes 16-31 so the
    // whole wave issues one fully coalesced 256B b64 store.
    float t2 = __shfl(D[2], lane & 15, 32);
    float t3 = __shfl(D[3], lane & 15, 32);
    v2f o;
    o.x = (lane < 16) ? D[0] : t2;   // cA at even feature slot
    o.y = (lane < 16) ? D[1] : t3;   // cD at odd feature slot

    // out[b, i, 2*f0 + 2*l] for lane l covers 2*f0 .. 2*f0+63 contiguously
    float* op = ob + (size_t)i * (2 * F) + 2 * f0 + 2 * lane;
    __builtin_nontemporal_store(o, (v2f*)op);
  }
}

extern "C" void kernel_launch(void* const* d_in, const int* in_sizes, int n_in,
                              void* d_out, int out_size, void* d_ws, size_t ws_size,
                              hipStream_t stream) {
  const float* x   = (const float*)d_in[0];
  float*       out = (float*)d_out;
  dim3 grid(BDIM * PAIRS / ROWS);   // 16384 blocks
  dim3 block(256);                  // 8 wave32 per block, one pair-row each iter
  hipLaunchKernelGGL(haar_wmma_kernel, grid, block, 0, stream, x, out);
}